// Network_79139067396324
// MI455X (gfx1250) — compile-verified
//
#include <hip/hip_runtime.h>
#include <hip/hip_bf16.h>

#define NN 1000
#define NP 1008
#define TT 8192
#define NWG 63
#define CHUNKS 250                        // K chunks of 4: 250*4 = 1000
#define PAIRS  125                        // chunk pairs (2 chunks per b128 load)
#define TILE_FLOATS (NWG * CHUNKS * 64)   // 1,008,000 floats = 4.03 MB

typedef float v2f __attribute__((ext_vector_type(2)));
typedef float v4f __attribute__((ext_vector_type(4)));
typedef float v8f __attribute__((ext_vector_type(8)));

// ---------------------------------------------------------------------------
// Setup kernel 1: s_row[i] = sum_j (ew[i,j] + ew[j,i]) * ea[i,j]
// ---------------------------------------------------------------------------
__global__ __launch_bounds__(256) void rowsum_kernel(const float* __restrict__ ew,
                                                     const float* __restrict__ ea,
                                                     float* __restrict__ srow) {
    __shared__ float red[256];
    const int i = blockIdx.x;
    const int tid = threadIdx.x;
    float p = 0.f;
    for (int j = tid; j < NN; j += 256)
        p += (ew[i * NN + j] + ew[j * NN + i]) * ea[i * NN + j];
    red[tid] = p;
    __syncthreads();
    for (int s2 = 128; s2 > 0; s2 >>= 1) {
        if (tid < s2) red[tid] += red[tid + s2];
        __syncthreads();
    }
    if (tid == 0) srow[i] = red[0];
}

// ---------------------------------------------------------------------------
// Setup kernel 2: build A in WMMA f32 16x16x4 A-fragment layout, pair-tiled so
// each lane's fragments for chunks (2g, 2g+1) are 16B-contiguous:
//   At[blk*16000 + g*128 + lane*4 + sub*2 + comp]
//   c = 2g + sub, row = blk*16 + (lane&15), k = 4c + ((lane>>4)<<1) + comp
//   A(row,k) = signs[k]*cw*ca - S(row,k) + (row==k ? srow[row] : 0)
// ---------------------------------------------------------------------------
__global__ __launch_bounds__(256) void build_kernel(const float* __restrict__ cw,
                                                    const float* __restrict__ ew,
                                                    const float* __restrict__ ca,
                                                    const float* __restrict__ ea,
                                                    const float* __restrict__ signs,
                                                    const float* __restrict__ srow,
                                                    float* __restrict__ At) {
    const int idx = blockIdx.x * 256 + threadIdx.x;
    if (idx >= TILE_FLOATS) return;
    const int blk  = idx / (PAIRS * 128);
    const int rem  = idx % (PAIRS * 128);
    const int g    = rem / 128;
    const int q    = rem % 128;
    const int lane = q >> 2;
    const int r2   = q & 3;
    const int sub  = r2 >> 1;
    const int comp = r2 & 1;
    const int c    = 2 * g + sub;
    const int row  = blk * 16 + (lane & 15);
    const int k    = 4 * c + ((lane >> 4) << 1) + comp;
    float v = 0.f;
    if (row < NN && k < NN) {
        const float Sij = (ew[row * NN + k] + ew[k * NN + row]) * ea[row * NN + k];
        v = signs[k] * cw[row * NN + k] * ca[row * NN + k] - Sij;
        if (row == k) v += srow[row];
    }
    At[idx] = v;
}

// ---------------------------------------------------------------------------
// Setup kernel 3: init ping-pong x buffers + global barrier counter.
// ---------------------------------------------------------------------------
__global__ void init_kernel(const float* __restrict__ x0,
                            float* __restrict__ xb0, float* __restrict__ xb1,
                            unsigned* __restrict__ bar) {
    const int i = blockIdx.x * blockDim.x + threadIdx.x;
    if (i == 0) *bar = 0u;
    if (i < NP) {
        xb0[i] = (i < NN) ? x0[i] : 0.f;
        xb1[i] = 0.f;
    }
}

// ---------------------------------------------------------------------------
// Main persistent kernel: 63 WGs x 256 threads. WG wg owns rows [wg*16,wg*16+16).
// Per step: stage x to LDS, pair-split software-pipelined WMMA matvec across
// 8 waves, ds_add_f32 reduce, pointwise update + NT output, device barrier.
// ---------------------------------------------------------------------------
__global__ __launch_bounds__(256) void sim_kernel(const float* __restrict__ s,
                                                  const float* __restrict__ ntau,
                                                  const float* __restrict__ ctau,
                                                  const float* __restrict__ cshift,
                                                  const float* __restrict__ bias,
                                                  const float* __restrict__ x0,
                                                  const float* __restrict__ At,
                                                  float* __restrict__ xb0,
                                                  float* __restrict__ xb1,
                                                  unsigned* __restrict__ bar,
                                                  float* __restrict__ out) {
    __shared__ __align__(16) float xs[NP];
    __shared__ float ni[16];

    const int tid  = threadIdx.x;
    const int wg   = blockIdx.x;
    const int lane = tid & 31;
    const int wave = tid >> 5;
    const int row  = wg * 16 + tid;                 // meaningful for tid < 16
    const bool owner = (tid < 16) && (row < NN);

    // Branch-free B-fragment helpers: only lanes 0 and 16 contribute (column 0).
    const float lmask = ((lane & 15) == 0) ? 1.0f : 0.0f;
    const int   koff  = (lane >> 4) << 1;           // 0 for lanes 0-15, 2 for 16-31

    float freg = 0.f, invn = 0.f, invc = 0.f, brow = 0.f, shiftv = 0.f;
    if (owner) {
        freg   = x0[row];
        invn   = 0.2f / fmaxf(ntau[row], 0.2f);
        invc   = 0.2f / fmaxf(ctau[row], 0.2f);
        brow   = bias[row];
        shiftv = cshift[row];
    }

    // Pair range for this wave: waves 0..6 get 16 pairs, wave 7 gets 13.
    const int p0 = wave * 16;
    const int p1 = (p0 + 16 < PAIRS) ? (p0 + 16) : PAIRS;
    const v4f* __restrict__ Abase =
        (const v4f*)At + (size_t)wg * PAIRS * 32 + (size_t)p0 * 32 + lane;

    for (int t = 0; t < TT; ++t) {
        const float* __restrict__ xc = (t & 1) ? xb1 : xb0;
        float* __restrict__ xn       = (t & 1) ? xb0 : xb1;

        // Stage full x vector into LDS (b128, coalesced); zero reduction slot.
        if (tid < NP / 4) ((v4f*)xs)[tid] = ((const v4f*)xc)[tid];
        if (tid < 16) ni[tid] = 0.f;
        __syncthreads();

        // Software-pipelined WMMA matvec: prefetch g+1 while computing g so the
        // WMMA pair only needs partial s_wait counts in steady state.
        v8f acc = {0.f, 0.f, 0.f, 0.f, 0.f, 0.f, 0.f, 0.f};
        const v4f* __restrict__ Ap = Abase;
        v4f a4  = *Ap;                                        // global_load_b128
        v2f xv0 = *(const v2f*)(xs + 8 * p0 + koff);          // ds_load (broadcast)
        v2f xv1 = *(const v2f*)(xs + 8 * p0 + 4 + koff);
        for (int g = p0; g < p1 - 1; ++g) {
            Ap += 32;
            const v4f an  = *Ap;                                    // prefetch next
            const v2f xn0 = *(const v2f*)(xs + 8 * (g + 1) + koff);
            const v2f xn1 = *(const v2f*)(xs + 8 * (g + 1) + 4 + koff);
            v2f a0; a0.x = a4.x; a0.y = a4.y;
            v2f a1; a1.x = a4.z; a1.y = a4.w;
            v2f b0; b0.x = xv0.x * lmask; b0.y = xv0.y * lmask;
            v2f b1; b1.x = xv1.x * lmask; b1.y = xv1.y * lmask;
            acc = __builtin_amdgcn_wmma_f32_16x16x4_f32(
                false, a0, false, b0, (short)0, acc, false, false);
            acc = __builtin_amdgcn_wmma_f32_16x16x4_f32(
                false, a1, false, b1, (short)0, acc, false, false);
            a4 = an; xv0 = xn0; xv1 = xn1;
        }
        {   // epilogue: last pair
            v2f a0; a0.x = a4.x; a0.y = a4.y;
            v2f a1; a1.x = a4.z; a1.y = a4.w;
            v2f b0; b0.x = xv0.x * lmask; b0.y = xv0.y * lmask;
            v2f b1; b1.x = xv1.x * lmask; b1.y = xv1.y * lmask;
            acc = __builtin_amdgcn_wmma_f32_16x16x4_f32(
                false, a0, false, b0, (short)0, acc, false, false);
            acc = __builtin_amdgcn_wmma_f32_16x16x4_f32(
                false, a1, false, b1, (short)0, acc, false, false);
        }

        // D column 0: lane 0 -> rows M=0..7, lane 16 -> rows M=8..15.
        if ((lane & 15) == 0) {
            const int base = (lane >> 4) * 8;
            #pragma unroll
            for (int r = 0; r < 8; ++r) atomicAdd(&ni[base + r], acc[r]);
        }
        __syncthreads();

        // Pointwise recurrence + non-temporal output streams.
        if (owner) {
            const float niv = ni[tid];
            const float xo  = xs[row];
            const float st  = __builtin_nontemporal_load(&s[(size_t)t * NN + row]);
            const float pre = niv + brow + st;
            const float xt  = xo + invn * (fmaxf(pre, 0.f) - xo);
            const float ft  = freg + invc * (xo - freg);
            const size_t base = (size_t)t * NN + row;
            __builtin_nontemporal_store(xt,          &out[base]);                       // x_slow
            __builtin_nontemporal_store(ft + shiftv, &out[(size_t)TT * NN + base]);     // f_slow
            __builtin_nontemporal_store(niv,         &out[(size_t)2 * TT * NN + base]); // neural_in
            __builtin_nontemporal_store(st,          &out[(size_t)3 * TT * NN + base]); // sensory_in
            xn[row] = xt;
            freg = ft;
        } else if (tid < 16) {
            xn[row] = 0.f;                    // keep pad rows zeroed
        }

        // Device-wide barrier (all 63 tiny WGs co-resident).
        __threadfence();
        __syncthreads();
        if (tid == 0) {
            __hip_atomic_fetch_add(bar, 1u, __ATOMIC_RELEASE, __HIP_MEMORY_SCOPE_AGENT);
            const unsigned target = (unsigned)(t + 1) * NWG;
            while (__hip_atomic_load(bar, __ATOMIC_ACQUIRE, __HIP_MEMORY_SCOPE_AGENT) < target) {
                __builtin_amdgcn_s_sleep(1);
            }
        }
        __syncthreads();
    }
}

// ---------------------------------------------------------------------------
extern "C" void kernel_launch(void* const* d_in, const int* in_sizes, int n_in,
                              void* d_out, int out_size, void* d_ws, size_t ws_size,
                              hipStream_t stream) {
    const float* s        = (const float*)d_in[0];
    const float* chem_w   = (const float*)d_in[1];
    const float* elec_w   = (const float*)d_in[2];
    const float* chem_adj = (const float*)d_in[3];
    const float* elec_adj = (const float*)d_in[4];
    const float* signs    = (const float*)d_in[5];
    const float* ntau     = (const float*)d_in[6];
    const float* ctau     = (const float*)d_in[7];
    const float* cshift   = (const float*)d_in[8];
    const float* bias     = (const float*)d_in[9];
    const float* x0       = (const float*)d_in[10];
    float* out = (float*)d_out;

    // Workspace layout (floats), all 16B-aligned
    float* ws   = (float*)d_ws;
    float* At   = ws;                                  // TILE_FLOATS
    float* srow = ws + TILE_FLOATS;                    // NN (padded to NP)
    float* xbA  = srow + NP;                           // NP
    float* xbB  = xbA + NP;                            // NP
    unsigned* bar = (unsigned*)(xbB + NP);             // 1

    rowsum_kernel<<<NN, 256, 0, stream>>>(elec_w, elec_adj, srow);
    build_kernel<<<(TILE_FLOATS + 255) / 256, 256, 0, stream>>>(
        chem_w, elec_w, chem_adj, elec_adj, signs, srow, At);
    init_kernel<<<(NP + 255) / 256, 256, 0, stream>>>(x0, xbA, xbB, bar);
    sim_kernel<<<NWG, 256, 0, stream>>>(s, ntau, ctau, cshift, bias, x0,
                                        At, xbA, xbB, bar, out);
    (void)in_sizes; (void)n_in; (void)out_size; (void)ws_size;
}